// WNNLUTLayer_47828755808728
// MI455X (gfx1250) — compile-verified
//
#include <hip/hip_runtime.h>
#include <cstdint>

// Problem constants (match reference)
#define B_DIM     1024
#define IN_BITS   4096
#define NUM_LUTS  16384
#define K_BITS    6
#define WORDS_PER_ROW (IN_BITS / 32)   // 128 u32 words per packed batch row

// Tiling for the lookup kernel
#define LT      128   // LUTs per block (lane <-> LUT, 4 waves)
#define BCHUNK  64    // batch rows per block
#define TPB     128   // threads per block (4 wave32)

// ---------------------------------------------------------------------------
// Kernel 1: binarize x_bits and pack 32 bits/word via wave32 ballot.
// packed[b*128 + w] bit j = (x_bits[b, 32*w + j] > 0.5)
// ---------------------------------------------------------------------------
__global__ __launch_bounds__(256) void wnn_pack_bits(const float* __restrict__ x,
                                                     uint32_t* __restrict__ packed) {
    uint32_t tid = blockIdx.x * 256u + threadIdx.x;   // exact: B*IN_BITS = 16384*256
    float v = x[tid];
    uint32_t m;
#if __has_builtin(__builtin_amdgcn_ballot_w32)
    m = __builtin_amdgcn_ballot_w32(v > 0.5f);
#else
    m = (uint32_t)__ballot(v > 0.5f);
#endif
    if ((threadIdx.x & 31u) == 0u) packed[tid >> 5] = m;
}

// ---------------------------------------------------------------------------
// Kernel 2: per-block 128 LUTs x 64 batch rows.
//  - table tile staged in LDS, rows padded to 65 dwords (bank scrambling)
//  - packed bit-row (512B) double-buffered in LDS via async global->LDS DMA
//  - lane <-> LUT mapping => coalesced 128B-per-wave output stores
// ---------------------------------------------------------------------------
__global__ __launch_bounds__(TPB) void wnn_lut_lookup(const float* __restrict__ table,
                                                      const int* __restrict__ conn,
                                                      const uint32_t* __restrict__ packed,
                                                      float* __restrict__ out) {
    __shared__ float    tableS[LT * 65];              // 33,280 B
    __shared__ uint32_t packedS[2][WORDS_PER_ROW];    //  1,024 B

    const uint32_t tid     = threadIdx.x;             // 0..127 : local LUT id
    const uint32_t lutBase = blockIdx.x * LT;
    const uint32_t bBase   = blockIdx.y * BCHUNK;
    const uint32_t myLut   = lutBase + tid;

    // Per-lane connection data: word index + bit shift, K=6 pairs, loaded once.
    uint32_t w[K_BITS], sh[K_BITS];
#pragma unroll
    for (int k = 0; k < K_BITS; ++k) {
        uint32_t c = (uint32_t)conn[(size_t)myLut * K_BITS + k];
        w[k]  = c >> 5;     // < 128
        sh[k] = c & 31u;
    }

    // Wave 0: kick off async staging of the first packed row (32 lanes x b128 = 512B).
    if (tid < 32) {
        const uint32_t* g = packed + (size_t)bBase * WORDS_PER_ROW + tid * 4u;
        uint32_t l = (uint32_t)(uintptr_t)&packedS[0][tid * 4u];
        asm volatile("global_load_async_to_lds_b128 %0, %1, off"
                     :: "v"(l), "v"(g) : "memory");
    }

    // Stage the 128x64 table tile into LDS (coalesced loads, padded rows).
    for (uint32_t j = tid; j < LT * 64u; j += TPB) {
        tableS[(j >> 6) * 65u + (j & 63u)] = table[(size_t)lutBase * 64u + j];
    }

    for (uint32_t bi = 0; bi < BCHUNK; ++bi) {
        const uint32_t buf = bi & 1u;

        // All waves done reading buf^1 (prev iteration) and table is staged.
        __syncthreads();

        if (tid < 32) {
            if (bi + 1 < BCHUNK) {
                // Prefetch next packed row into the other buffer, then make sure
                // the row for THIS iteration (issued one iteration ago) landed.
                const uint32_t* g = packed + (size_t)(bBase + bi + 1) * WORDS_PER_ROW + tid * 4u;
                uint32_t l = (uint32_t)(uintptr_t)&packedS[buf ^ 1u][tid * 4u];
                asm volatile("global_load_async_to_lds_b128 %0, %1, off"
                             :: "v"(l), "v"(g) : "memory");
                asm volatile("s_wait_asynccnt 0x1" ::: "memory");
            } else {
                asm volatile("s_wait_asynccnt 0x0" ::: "memory");
            }
        }
        __syncthreads();  // packedS[buf] now valid for all waves

        // Gather 6 bits -> 6-bit LUT index (per-lane DS reads from 128-word row).
        uint32_t idx = 0;
#pragma unroll
        for (int k = 0; k < K_BITS; ++k) {
            idx = (idx << 1) | ((packedS[buf][w[k]] >> sh[k]) & 1u);
        }

        // Table lookup (padded stride 65 spreads banks) + fast sigmoid.
        float v = tableS[tid * 65u + idx];
        float e = __builtin_amdgcn_exp2f(v * -1.4426950408889634f);
        float r = __builtin_amdgcn_rcpf(1.0f + e);

        // Coalesced: each wave writes 32 consecutive floats of out[b, :].
        out[(size_t)(bBase + bi) * NUM_LUTS + myLut] = r;
    }
}

// ---------------------------------------------------------------------------
extern "C" void kernel_launch(void* const* d_in, const int* in_sizes, int n_in,
                              void* d_out, int out_size, void* d_ws, size_t ws_size,
                              hipStream_t stream) {
    (void)in_sizes; (void)n_in; (void)out_size; (void)ws_size;

    const float*    x     = (const float*)d_in[0];      // (1024, 4096) f32
    const float*    table = (const float*)d_in[1];      // (16384, 64) f32
    const int*      conn  = (const int*)d_in[2];        // (16384, 6) int
    float*          out   = (float*)d_out;              // (1024, 16384) f32
    uint32_t*       packed = (uint32_t*)d_ws;           // 1024*128*4 = 512 KB scratch

    // Pack: 4,194,304 bits -> 131,072 u32 words.
    wnn_pack_bits<<<(B_DIM * IN_BITS) / 256, 256, 0, stream>>>(x, packed);

    // Lookup: 128 LUT tiles x 16 batch chunks = 2048 blocks.
    dim3 grid(NUM_LUTS / LT, B_DIM / BCHUNK);
    wnn_lut_lookup<<<grid, TPB, 0, stream>>>(table, conn, packed, out);
}